// GCN_77498389889734
// MI455X (gfx1250) — compile-verified
//
#include <hip/hip_runtime.h>
#include <math.h>

// ---------------------------------------------------------------------------
// NNConv GNN (3 layers) for MI455X / gfx1250, wave32.
//
// msg_e = e*(x[src]@Wm) + x[src]@Bm is linear in x[src] =>
//   agg = (scatter_add(e*x[src])) @ Wm + (scatter_add(x[src])) @ Bm
// Phase A (per layer): S1[d] += e*x[s], S0[d] += x[s]   (L2-resident atomics)
// Phase B (per layer): out = [S1|S0|x] @ [Wm;Bm;Wr] + b  via V_WMMA_F32_16X16X4_F32
// Layer 3 (d_out=4) fused with log_softmax in a scalar kernel.
//
// B operands are staged in LDS as K-pairs ({W[k][n],W[k+1][n]} contiguous) so
// every WMMA B-operand is a single aligned ds_load_b64 (no repack moves), with
// pair-row stride 65 to de-conflict the two lane halves' LDS banks.
// ---------------------------------------------------------------------------

typedef float v2f __attribute__((ext_vector_type(2)));
typedef float v8f __attribute__((ext_vector_type(8)));

#define D 64              // feature width of layers 1/2 (D_IN == D_HID == 64)
#define PSTRIDE 65        // padded row stride (in float2) of the K-pair layout
#define SEGF2  (32 * PSTRIDE)   // float2 elements per 64-K weight segment

// ---------------- zero fill (S0 and S1 are contiguous) ----------------------
__global__ __launch_bounds__(256) void nnconv_zero(float* __restrict__ p, int n) {
  int i = blockIdx.x * 256 + threadIdx.x;
  if (i < n) p[i] = 0.0f;
}

// ---------------- edge scatter: one wave32 per edge -------------------------
// Each lane handles 2 of the 64 features (float2 = fully coalesced 256B row).
__global__ __launch_bounds__(256) void nnconv_scatter(
    const float* __restrict__ X, const int* __restrict__ src,
    const int* __restrict__ dst, const float* __restrict__ eattr,
    float* __restrict__ S0, float* __restrict__ S1, int E) {
  int warp = (blockIdx.x * 256 + threadIdx.x) >> 5;
  int lane = threadIdx.x & 31;
  if (warp >= E) return;                       // wave-uniform
  int s = src[warp];
  int d = dst[warp];
  float e = eattr[warp];
  const float2* xr = (const float2*)(X + (size_t)s * D);
  float2 v = xr[lane];
  float* p0 = S0 + (size_t)d * D + lane * 2;
  float* p1 = S1 + (size_t)d * D + lane * 2;
  __hip_atomic_fetch_add(p0 + 0, v.x,     __ATOMIC_RELAXED, __HIP_MEMORY_SCOPE_AGENT);
  __hip_atomic_fetch_add(p0 + 1, v.y,     __ATOMIC_RELAXED, __HIP_MEMORY_SCOPE_AGENT);
  __hip_atomic_fetch_add(p1 + 0, e * v.x, __ATOMIC_RELAXED, __HIP_MEMORY_SCOPE_AGENT);
  __hip_atomic_fetch_add(p1 + 1, e * v.y, __ATOMIC_RELAXED, __HIP_MEMORY_SCOPE_AGENT);
}

// ---------------- WMMA GEMM: [rows x 192] @ [192 x 64] ----------------------
// A 16x4 f32 layout: lanes 0-15 -> (M=lane, K=k0,k0+1); lanes 16-31 -> K=k0+2,k0+3.
// B 4x16 mirrors A: lanes 0-15 carry K-pair (k0,k0+1), lanes 16-31 (k0+2,k0+3),
// column n0+m.  Wseg holds K-pairs: Wseg[p*PSTRIDE + n] = {W[2p][n], W[2p+1][n]}.
__device__ __forceinline__ void wmma_seg64(const v2f* __restrict__ Arow2,
                                           const v2f* __restrict__ Wseg,
                                           int phalf, int m, v8f* acc) {
#pragma unroll
  for (int kk = 0; kk < 16; ++kk) {
    v2f a = Arow2[kk * 2 + phalf];           // {A[row][k0], A[row][k0+1]}
    const int p = kk * 2 + phalf;            // K-pair index for this lane half
#pragma unroll
    for (int nt = 0; nt < 4; ++nt) {
      v2f b = Wseg[p * PSTRIDE + nt * 16 + m];
      acc[nt] = __builtin_amdgcn_wmma_f32_16x16x4_f32(
          false, a, false, b, (short)0, acc[nt], false, false);
    }
  }
}

// 256 threads = 8 waves; each wave produces a 16x64 output tile (128 rows/block).
__global__ __launch_bounds__(256) void nnconv_mm64(
    const float* __restrict__ S1, const float* __restrict__ S0,
    const float* __restrict__ X,  const float* __restrict__ We,
    const float* __restrict__ Be, const float* __restrict__ Wr,
    const float* __restrict__ bias, float* __restrict__ out,
    int N, int doRelu) {
  __shared__ v2f Wlds[3 * SEGF2];            // ~49 KB, K-pair layout per segment
  float* wf = (float*)Wlds;
  const int tid = threadIdx.x;
  for (int idx = tid; idx < 3 * D * D; idx += 256) {
    const int seg = idx >> 12;               // 0:Wm 1:Bm 2:Wr
    const int rem = idx & 4095;
    const int k = rem >> 6, n = rem & 63;
    const float* p = (seg == 0) ? We : (seg == 1) ? Be : Wr;
    wf[(seg * SEGF2 + (k >> 1) * PSTRIDE + n) * 2 + (k & 1)] = p[rem];
  }
  __syncthreads();

  const int lane = tid & 31;
  const int wave = tid >> 5;
  const int row_base = blockIdx.x * 128 + wave * 16;
  if (row_base >= N) return;                 // wave-uniform: EXEC stays all-1s

  const int m = lane & 15;
  const int phalf = lane >> 4;               // 0: K-pair (k0,k0+1), 1: (k0+2,k0+3)
  const int row = row_base + m;
  const int rowc = (row < N) ? row : (N - 1);  // clamp reads; stores guarded

  v8f acc[4] = {};                           // 4 tiles of 16 cols each (N=64)
  wmma_seg64((const v2f*)(S1 + (size_t)rowc * D), Wlds,             phalf, m, acc);
  wmma_seg64((const v2f*)(S0 + (size_t)rowc * D), Wlds + SEGF2,     phalf, m, acc);
  wmma_seg64((const v2f*)(X  + (size_t)rowc * D), Wlds + 2 * SEGF2, phalf, m, acc);

  // C/D layout: VGPR j -> row j (lanes 0-15) / row j+8 (lanes 16-31), col = m.
  const int rstore = row_base + (phalf << 3);
#pragma unroll
  for (int nt = 0; nt < 4; ++nt) {
#pragma unroll
    for (int j = 0; j < 8; ++j) {
      int r = rstore + j;
      if (r < N) {
        int col = nt * 16 + m;
        float v = acc[nt][j] + bias[col];
        if (doRelu) v = v > 0.0f ? v : 0.0f;
        out[(size_t)r * D + col] = v;
      }
    }
  }
}

// ---------------- layer 3 (d_out=4) + log_softmax, fused --------------------
__global__ __launch_bounds__(256) void nnconv_l3(
    const float* __restrict__ S1, const float* __restrict__ S0,
    const float* __restrict__ X,  const float* __restrict__ We,
    const float* __restrict__ Be, const float* __restrict__ Wr,
    const float* __restrict__ bias, float* __restrict__ out, int N) {
  __shared__ float w[3 * D * 4];             // 3 KB
  for (int idx = threadIdx.x; idx < 3 * D * 4; idx += 256) {
    const float* p = (idx < 256) ? We : (idx < 512) ? Be : Wr;
    w[idx] = p[idx & 255];
  }
  __syncthreads();

  int i = blockIdx.x * 256 + threadIdx.x;
  if (i >= N) return;
  const float* r1 = S1 + (size_t)i * D;
  const float* r0 = S0 + (size_t)i * D;
  const float* rx = X  + (size_t)i * D;
  float acc[4] = {bias[0], bias[1], bias[2], bias[3]};
#pragma unroll 4
  for (int k = 0; k < D; ++k) {
    float a1 = r1[k], a0 = r0[k], ax = rx[k];
#pragma unroll
    for (int o = 0; o < 4; ++o)
      acc[o] += a1 * w[k * 4 + o] + a0 * w[256 + k * 4 + o] + ax * w[512 + k * 4 + o];
  }
  float mx = fmaxf(fmaxf(acc[0], acc[1]), fmaxf(acc[2], acc[3]));
  float s = 0.0f;
#pragma unroll
  for (int o = 0; o < 4; ++o) s += __expf(acc[o] - mx);
  float lse = mx + __logf(s);
#pragma unroll
  for (int o = 0; o < 4; ++o) out[(size_t)i * 4 + o] = acc[o] - lse;
}

// ---------------------------------------------------------------------------
extern "C" void kernel_launch(void* const* d_in, const int* in_sizes, int n_in,
                              void* d_out, int out_size, void* d_ws, size_t ws_size,
                              hipStream_t stream) {
  (void)n_in; (void)out_size; (void)ws_size;
  const float* x   = (const float*)d_in[0];
  const int*   ei  = (const int*)d_in[1];
  const float* ea  = (const float*)d_in[2];
  const float* We1 = (const float*)d_in[3];
  const float* be1 = (const float*)d_in[4];
  const float* Wr1 = (const float*)d_in[5];
  const float* b1  = (const float*)d_in[6];
  const float* We2 = (const float*)d_in[7];
  const float* be2 = (const float*)d_in[8];
  const float* Wr2 = (const float*)d_in[9];
  const float* b2  = (const float*)d_in[10];
  const float* We3 = (const float*)d_in[11];
  const float* be3 = (const float*)d_in[12];
  const float* Wr3 = (const float*)d_in[13];
  const float* b3  = (const float*)d_in[14];

  const int N = in_sizes[0] / D;             // 100000
  const int E = in_sizes[2];                 // 1600000
  const int* src = ei;                       // edge_index row 0
  const int* dst = ei + E;                   // edge_index row 1

  // workspace layout: S0 | S1 | h1 | h2  (each N*64 f32) = ~102 MB
  float* S0  = (float*)d_ws;
  float* S1  = S0 + (size_t)N * D;
  float* h1  = S1 + (size_t)N * D;
  float* h2  = h1 + (size_t)N * D;
  float* out = (float*)d_out;

  const int zn = 2 * N * D;                  // S0+S1 contiguous
  dim3 blk(256);
  dim3 gz((zn + 255) / 256);
  dim3 gs((unsigned)(((long long)E * 32 + 255) / 256));
  dim3 gm((N + 127) / 128);
  dim3 g3((N + 255) / 256);

  // layer 1: x -> h1
  nnconv_zero<<<gz, blk, 0, stream>>>(S0, zn);
  nnconv_scatter<<<gs, blk, 0, stream>>>(x, src, dst, ea, S0, S1, E);
  nnconv_mm64<<<gm, blk, 0, stream>>>(S1, S0, x, We1, be1, Wr1, b1, h1, N, 1);

  // layer 2: h1 -> h2
  nnconv_zero<<<gz, blk, 0, stream>>>(S0, zn);
  nnconv_scatter<<<gs, blk, 0, stream>>>(h1, src, dst, ea, S0, S1, E);
  nnconv_mm64<<<gm, blk, 0, stream>>>(S1, S0, h1, We2, be2, Wr2, b2, h2, N, 1);

  // layer 3: h2 -> log_softmax out
  nnconv_zero<<<gz, blk, 0, stream>>>(S0, zn);
  nnconv_scatter<<<gs, blk, 0, stream>>>(h2, src, dst, ea, S0, S1, E);
  nnconv_l3<<<g3, blk, 0, stream>>>(S1, S0, h2, We3, be3, Wr3, b3, out, N);
}